// CSA4Rec_Encoder_8160437862431
// MI455X (gfx1250) — compile-verified
//
#include <hip/hip_runtime.h>
#include <hip/hip_bf16.h>

// ---------------------------------------------------------------------------
// SimGCL-style graph encoder for gfx1250 (MI455X).
//   1) init: ego0 = concat(user_emb, item_emb); acc = 0; counts = 0
//   2) CSR build: histogram -> 1-WG scan -> scatter. Scatter packs
//      (local_row = row & 15) with the column index into one word and stores
//      {packed, val} as int2 -> single b64 load per edge, no row search.
//   3) per layer: fused CSR-SpMM via V_WMMA_F32_16X16X4_F32 (16-row blocks),
//      + sign(y)*normalize(noise)*eps perturbation, + acc += y/3,
//      + emb_cl copy at layer 0. Epilogue goes through an LDS tile so every
//      global access is a fully-coalesced b128.
// Entire node table (41 MB) is L2-resident on MI455X (192 MB L2), so the CSR
// SpMM is L2-gather bound and atomic-free; WMMA redundancy (~8 GFLOP) is free.
// ---------------------------------------------------------------------------

#define N_USERS 100000
#define N_ITEMS 60000
#define N_NODES (N_USERS + N_ITEMS)
#define D 64
#define NNZ 1280000
#define EPS 0.1f
#define WAVES_PER_BLK 4
#define COL_BITS 20           // N_NODES < 2^20; lrow lives in bits [24:20]
#define COL_MASK ((1 << COL_BITS) - 1)

static_assert(N_NODES % 16 == 0, "block tiling requires N_NODES % 16 == 0");
static_assert(N_NODES <= (1 << COL_BITS), "column index must fit in COL_BITS");

typedef __attribute__((ext_vector_type(2))) float v2f;
typedef __attribute__((ext_vector_type(8))) float v8f;

// ---------------------------------------------------------------------------
__global__ __launch_bounds__(256)
void init_kernel(const float* __restrict__ ue, const float* __restrict__ ie,
                 float* __restrict__ ego0, float* __restrict__ acc,
                 unsigned* __restrict__ counts) {
  int i = blockIdx.x * 256 + threadIdx.x;
  const int total = N_NODES * D;
  if (i < total) {
    ego0[i] = (i < N_USERS * D) ? ue[i] : ie[i - N_USERS * D];
    acc[i] = 0.0f;
  }
  if (i < N_NODES) counts[i] = 0u;
}

// ---------------------------------------------------------------------------
__global__ __launch_bounds__(256)
void hist_kernel(const int* __restrict__ rows, unsigned* __restrict__ counts) {
  int e = blockIdx.x * 256 + threadIdx.x;
  if (e < NNZ) atomicAdd(&counts[rows[e]], 1u);
}

// ---------------------------------------------------------------------------
// Single-workgroup chunked Hillis-Steele exclusive scan over 160k counts.
// Tiny cost (~160k adds) relative to the SpMM; keeps the pipeline simple.
__global__ __launch_bounds__(1024)
void scan_kernel(const unsigned* __restrict__ counts,
                 unsigned* __restrict__ row_ptr, unsigned* __restrict__ cursor) {
  __shared__ unsigned buf[1024];
  __shared__ unsigned s_carry;
  const int t = threadIdx.x;
  if (t == 0) s_carry = 0u;
  __syncthreads();
  for (int base = 0; base < N_NODES; base += 1024) {
    unsigned carry_in = s_carry;
    unsigned v = (base + t < N_NODES) ? counts[base + t] : 0u;
    buf[t] = v;
    __syncthreads();
    for (int off = 1; off < 1024; off <<= 1) {
      unsigned u = (t >= off) ? buf[t - off] : 0u;
      __syncthreads();
      buf[t] += u;
      __syncthreads();
    }
    unsigned incl = buf[t];
    if (base + t < N_NODES) {
      unsigned e = carry_in + incl - v;
      row_ptr[base + t] = e;
      cursor[base + t] = e;
    }
    unsigned tot = buf[1023];
    __syncthreads();
    if (t == 0) s_carry = carry_in + tot;
    __syncthreads();
  }
  if (t == 0) row_ptr[N_NODES] = s_carry;  // == NNZ
}

// ---------------------------------------------------------------------------
// Sort edges by destination row; pack (lrow << COL_BITS) | col with the value
// as an int2 so the SpMM consumes one b64 load per edge.
__global__ __launch_bounds__(256)
void scatter_kernel(const int* __restrict__ rows, const int* __restrict__ cols,
                    const float* __restrict__ vals, unsigned* __restrict__ cursor,
                    int2* __restrict__ sedge) {
  int e = blockIdx.x * 256 + threadIdx.x;
  if (e < NNZ) {
    int r = rows[e];
    unsigned p = atomicAdd(&cursor[r], 1u);
    int2 ed;
    ed.x = ((r & 15) << COL_BITS) | cols[e];
    ed.y = __float_as_int(vals[e]);
    sedge[p] = ed;
  }
}

// ---------------------------------------------------------------------------
// One wave (32 lanes) per 16-row output block. D tiles: 4x v8f (16x16 f32).
// Edge groups of 32 (one b64 load/lane), consumed as 8 K-chunks of 4 edges:
// A = one-hot-scaled 16x4 (A[m,k]=val iff edge k targets local row m);
// B = gathered x rows (4x16 per tile) staged in LDS.
// A layout per CDNA5 ISA (32-bit 16x4): lanes 0-15 hold M=lane, VGPR0=K0,
// VGPR1=K1; lanes 16-31 hold M=lane-16, VGPR0=K2, VGPR1=K3. B (4x16) uses the
// symmetric transposed striping with N = lane%16.
__global__ __launch_bounds__(32 * WAVES_PER_BLK)
void spmm_wmma_kernel(const float* __restrict__ x,
                      const unsigned* __restrict__ row_ptr,
                      const int2* __restrict__ sedge,
                      const float* __restrict__ noise_k,
                      float* __restrict__ x_out,
                      float* __restrict__ acc,
                      float* __restrict__ embcl,
                      int write_cl) {
  __shared__ __align__(16) float s_x[WAVES_PER_BLK][4 * D];   // 4 gathered rows
  __shared__ __align__(16) float s_y[WAVES_PER_BLK][16 * D];  // D-tile staging

  const int lane = threadIdx.x & 31;
  const int wid = threadIdx.x >> 5;
  const int blk16 = blockIdx.x * WAVES_PER_BLK + wid;  // 16-row block id
  const int r0 = blk16 * 16;

  float* lds_x = s_x[wid];
  float* lds_y = s_y[wid];

  const unsigned span_s = row_ptr[r0];
  const unsigned span_e = row_ptr[r0 + 16];
  const unsigned nedges = span_e - span_s;
  const int ngroups = (int)((nedges + 31u) >> 5);

  v8f c[4] = {};  // 16x64 f32 accumulator (4 WMMA tiles)

  const int m = lane & 15;          // M (for A) / N (for B and C/D)
  const int kA = (lane >> 4) << 1;  // K pair handled by this half-wave

  for (int g = 0; g < ngroups; ++g) {
    const unsigned gbase = span_s + ((unsigned)g << 5);

    // 32 edges per group: one b64 load per lane; tail lanes get lrow=16
    // (outside 0..15) so their A entries are forced to zero.
    int mypacked = 16 << COL_BITS;
    float myval = 0.0f;
    {
      unsigned e = gbase + (unsigned)lane;
      if (e < span_e) {
        int2 ed = sedge[e];
        mypacked = ed.x;
        myval = __int_as_float(ed.y);
      }
    }
    const unsigned rem = span_e - gbase;
    const int nch = (int)(((rem < 32u ? rem : 32u) + 3u) >> 2);

    for (int ch = 0; ch < nch; ++ch) {  // wave-uniform trip count
      int colk[4];
      float valk[4];
      int lrowk[4];
#pragma unroll
      for (int k = 0; k < 4; ++k) {
        int src = ch * 4 + k;  // wave-uniform -> v_readlane
        int pk = __builtin_amdgcn_readlane(mypacked, src);
        valk[k] = __int_as_float(
            __builtin_amdgcn_readlane(__float_as_int(myval), src));
        colk[k] = pk & COL_MASK;
        lrowk[k] = pk >> COL_BITS;
      }

      // Stage the 4 gathered x rows (4 x 64 f32) into LDS: 64 float4 slots,
      // 2 per lane, fully coalesced 256B/row reads (L2-resident table).
#pragma unroll
      for (int rep = 0; rep < 2; ++rep) {
        int f = lane + rep * 32;  // 0..63
        int kr = f >> 4, e4 = f & 15;
        float4 v = ((const float4*)(x + (size_t)colk[kr] * D))[e4];
        ((float4*)lds_x)[f] = v;
      }
      asm volatile("s_wait_dscnt 0x0" ::: "memory");  // cross-lane visibility

      // A operand: one-hot-scaled selection matrix.
      v2f A;
      A.x = (lrowk[kA] == m) ? valk[kA] : 0.0f;
      A.y = (lrowk[kA + 1] == m) ? valk[kA + 1] : 0.0f;

#pragma unroll
      for (int t = 0; t < 4; ++t) {
        v2f B;
        B.x = lds_x[(kA)*D + t * 16 + m];
        B.y = lds_x[(kA + 1) * D + t * 16 + m];
        c[t] = __builtin_amdgcn_wmma_f32_16x16x4_f32(
            false, A, false, B, (short)0, c[t], false, false);
      }
    }
  }

  // Dump D tiles to LDS: C/D layout -> VGPR j holds M=j (lanes 0-15) / M=j+8
  // (lanes 16-31), N = lane%16. Conflict-free b32 stores.
#pragma unroll
  for (int t = 0; t < 4; ++t) {
#pragma unroll
    for (int j = 0; j < 8; ++j) {
      int lr = j + ((lane >> 4) << 3);
      lds_y[lr * D + t * 16 + m] = c[t][j];
    }
  }
  asm volatile("s_wait_dscnt 0x0" ::: "memory");

  // Coalesced epilogue: step i covers rows 2i (lanes 0-15) and 2i+1
  // (lanes 16-31), 512B contiguous per instruction. Each half-wave reduces
  // its row's noise norm with a 4-step butterfly (stays inside the half),
  // so noise is read exactly once and everything is b128.
  const float third = 1.0f / 3.0f;
#pragma unroll
  for (int i = 0; i < 8; ++i) {
    const int lr = i * 2 + (lane >> 4);  // local row 0..15
    const int c4 = lane & 15;            // float4 slot within the row
    const size_t base4 = (size_t)(r0 + lr) * (D / 4) + c4;  // float4 index

    float4 n4 = ((const float4*)noise_k)[base4];
    float ss = n4.x * n4.x + n4.y * n4.y + n4.z * n4.z + n4.w * n4.w;
#pragma unroll
    for (int off = 1; off <= 8; off <<= 1) ss += __shfl_xor(ss, off, 32);
    const float inv = EPS / fmaxf(sqrtf(ss), 1e-12f);

    float4 y4 = ((const float4*)(lds_y + lr * D))[c4];
    float4 o4;
    {
      float sg;
      sg = (y4.x > 0.f) ? 1.f : ((y4.x < 0.f) ? -1.f : 0.f);
      o4.x = y4.x + sg * n4.x * inv;
      sg = (y4.y > 0.f) ? 1.f : ((y4.y < 0.f) ? -1.f : 0.f);
      o4.y = y4.y + sg * n4.y * inv;
      sg = (y4.z > 0.f) ? 1.f : ((y4.z < 0.f) ? -1.f : 0.f);
      o4.z = y4.z + sg * n4.z * inv;
      sg = (y4.w > 0.f) ? 1.f : ((y4.w < 0.f) ? -1.f : 0.f);
      o4.w = y4.w + sg * n4.w * inv;
    }

    ((float4*)x_out)[base4] = o4;
    float4 a4 = ((const float4*)acc)[base4];  // each element owned by 1 lane
    a4.x += o4.x * third;
    a4.y += o4.y * third;
    a4.z += o4.z * third;
    a4.w += o4.w * third;
    ((float4*)acc)[base4] = a4;
    if (write_cl) ((float4*)embcl)[base4] = o4;
  }
}

// ---------------------------------------------------------------------------
extern "C" void kernel_launch(void* const* d_in, const int* in_sizes, int n_in,
                              void* d_out, int out_size, void* d_ws, size_t ws_size,
                              hipStream_t stream) {
  const float* user_emb = (const float*)d_in[0];
  const float* item_emb = (const float*)d_in[1];
  const float* adj_vals = (const float*)d_in[2];
  const float* noise    = (const float*)d_in[3];  // [3, N, D]
  const int*   adj_rows = (const int*)d_in[4];
  const int*   adj_cols = (const int*)d_in[5];

  float* acc   = (float*)d_out;                   // final  [N, D]
  float* embcl = acc + (size_t)N_NODES * D;       // emb_cl [N, D]

  // Bump-allocate workspace (~94 MB total).
  char* p = (char*)d_ws;
  auto alloc = [&](size_t bytes) -> char* {
    char* r = p;
    p += (bytes + 255) & ~(size_t)255;
    return r;
  };
  float*    ego0    = (float*)alloc(sizeof(float) * (size_t)N_NODES * D);
  float*    ego1    = (float*)alloc(sizeof(float) * (size_t)N_NODES * D);
  int2*     sedge   = (int2*)alloc(sizeof(int2) * (size_t)NNZ);
  unsigned* counts  = (unsigned*)alloc(sizeof(unsigned) * N_NODES);
  unsigned* row_ptr = (unsigned*)alloc(sizeof(unsigned) * (N_NODES + 1));
  unsigned* cursor  = (unsigned*)alloc(sizeof(unsigned) * N_NODES);

  init_kernel<<<(N_NODES * D + 255) / 256, 256, 0, stream>>>(
      user_emb, item_emb, ego0, acc, counts);
  hist_kernel<<<(NNZ + 255) / 256, 256, 0, stream>>>(adj_rows, counts);
  scan_kernel<<<1, 1024, 0, stream>>>(counts, row_ptr, cursor);
  scatter_kernel<<<(NNZ + 255) / 256, 256, 0, stream>>>(
      adj_rows, adj_cols, adj_vals, cursor, sedge);

  const size_t nd = (size_t)N_NODES * D;
  dim3 g((N_NODES / 16) / WAVES_PER_BLK);  // 2500 workgroups
  dim3 b(32 * WAVES_PER_BLK);              // 4 waves (wave32) each

  spmm_wmma_kernel<<<g, b, 0, stream>>>(ego0, row_ptr, sedge,
                                        noise + 0 * nd, ego1, acc, embcl, 1);
  spmm_wmma_kernel<<<g, b, 0, stream>>>(ego1, row_ptr, sedge,
                                        noise + 1 * nd, ego0, acc, embcl, 0);
  spmm_wmma_kernel<<<g, b, 0, stream>>>(ego0, row_ptr, sedge,
                                        noise + 2 * nd, ego1, acc, embcl, 0);
}